// GATLayer_14353780704047
// MI455X (gfx1250) — compile-verified
//
#include <hip/hip_runtime.h>

typedef __attribute__((ext_vector_type(2))) float v2f;
typedef __attribute__((ext_vector_type(8))) float v8f;

#define NN 50000
#define EE 600000
#define DD 128
#define HH 8
#define CC 16
#define HC 128   // H*C
#define NEG_SLOPE 0.2f

// ---------------------------------------------------------------------------
// Kernel 1: h = x @ W  (fp32 GEMM via V_WMMA_F32_16X16X4_F32)
// grid.x = 3125 row tiles (50000/16), block = 256 threads = 8 waves.
// Wave w computes the 16x16 tile at columns [16w, 16w+16) == head w.
// ---------------------------------------------------------------------------
__global__ __launch_bounds__(256) void gat_gemm_wmma(const float* __restrict__ x,
                                                     const float* __restrict__ W,
                                                     float* __restrict__ h) {
  const int lane = threadIdx.x & 31;
  const int wave = threadIdx.x >> 5;      // 0..7 -> column tile / head
  const int rowTile = blockIdx.x;         // 0..3124
  const int m = lane & 15;                // A row / C col index
  const int half = lane >> 4;             // lane half selects K pair / M half
  const int colBase = wave * 16;

  const float* __restrict__ xrow = x + (size_t)(rowTile * 16 + m) * DD;

  v8f acc = {};
  #pragma unroll
  for (int kb = 0; kb < DD; kb += 4) {
    const int k0 = kb + 2 * half;
    v2f a, b;
    // A 16x4 f32: lane m = row; VGPR v holds K = kb + 2*half + v
    a[0] = xrow[k0];
    a[1] = xrow[k0 + 1];
    // B 4x16 f32: lane n = col; VGPR v holds K-row kb + 2*half + v
    b[0] = W[(size_t)(k0)     * HC + colBase + m];
    b[1] = W[(size_t)(k0 + 1) * HC + colBase + m];
    acc = __builtin_amdgcn_wmma_f32_16x16x4_f32(
        /*neg_a=*/false, a, /*neg_b=*/false, b,
        /*c_mod=*/(short)0, acc, /*reuse_a=*/false, /*reuse_b=*/false);
  }

  // D 16x16 f32: VGPR v -> row (v + 8*half), lane -> col (lane&15)
  #pragma unroll
  for (int v = 0; v < 8; ++v) {
    h[(size_t)(rowTile * 16 + v + 8 * half) * HC + colBase + m] = acc[v];
  }
}

// ---------------------------------------------------------------------------
// Kernel 2: alpha_s[n,h] = <h[n,h,:], a_src[h,:]>, same for a_dst.
// One thread per (node, head).
// ---------------------------------------------------------------------------
__global__ void gat_alpha(const float* __restrict__ h,
                          const float* __restrict__ a_src,
                          const float* __restrict__ a_dst,
                          float* __restrict__ as_out,
                          float* __restrict__ ad_out) {
  const int t = blockIdx.x * blockDim.x + threadIdx.x;
  if (t >= NN * HH) return;
  const int n = t / HH;
  const int hh = t - n * HH;
  const float* __restrict__ hp = h + (size_t)n * HC + hh * CC;
  const float* __restrict__ s = a_src + hh * CC;
  const float* __restrict__ d = a_dst + hh * CC;
  float ss = 0.f, dd = 0.f;
  #pragma unroll
  for (int c = 0; c < CC; ++c) {
    const float v = hp[c];
    ss += v * s[c];
    dd += v * d[c];
  }
  as_out[t] = ss;
  ad_out[t] = dd;
}

// ---------------------------------------------------------------------------
// Kernel 3: init out <- bias (broadcast), emax <- encoded(-inf)=0, esum <- 0.
// ---------------------------------------------------------------------------
__global__ void gat_init(float* __restrict__ out, const float* __restrict__ bias,
                         unsigned* __restrict__ emax, float* __restrict__ esum) {
  const int t = blockIdx.x * blockDim.x + threadIdx.x;
  if (t < NN * HC) out[t] = bias[t & (HC - 1)];
  if (t < NN * HH) { emax[t] = 0u; esum[t] = 0.f; }
}

// Monotonic uint encoding for float atomic-max.
__device__ __forceinline__ unsigned fenc(float f) {
  const unsigned u = __float_as_uint(f);
  return (u & 0x80000000u) ? ~u : (u | 0x80000000u);
}
__device__ __forceinline__ float fdec(unsigned u) {
  return __uint_as_float((u & 0x80000000u) ? (u ^ 0x80000000u) : ~u);
}

// ---------------------------------------------------------------------------
// Kernel 4: per (edge,head) logits + LeakyReLU + segment max (atomicMax).
// Virtual edges i in [E, E+N) are the self loops.
// ---------------------------------------------------------------------------
__global__ void gat_edge_max(const int* __restrict__ ei,
                             const float* __restrict__ as,
                             const float* __restrict__ ad,
                             float* __restrict__ e,
                             unsigned* __restrict__ emax) {
  const int t = blockIdx.x * blockDim.x + threadIdx.x;
  if (t >= (EE + NN) * HH) return;
  const int i = t / HH;
  const int hh = t - i * HH;
  int s, d;
  if (i < EE) { s = ei[i]; d = ei[EE + i]; } else { s = d = i - EE; }
  float ev = as[s * HH + hh] + ad[d * HH + hh];
  ev = ev > 0.f ? ev : NEG_SLOPE * ev;
  e[t] = ev;
  atomicMax(&emax[d * HH + hh], fenc(ev));
}

// ---------------------------------------------------------------------------
// Kernel 5: e <- exp(e - emax[dst]); esum[dst] += e  (atomicAdd).
// ---------------------------------------------------------------------------
__global__ void gat_edge_exp(const int* __restrict__ ei,
                             float* __restrict__ e,
                             const unsigned* __restrict__ emax,
                             float* __restrict__ esum) {
  const int t = blockIdx.x * blockDim.x + threadIdx.x;
  if (t >= (EE + NN) * HH) return;
  const int i = t / HH;
  const int hh = t - i * HH;
  const int d = (i < EE) ? ei[EE + i] : (i - EE);
  const float m = fdec(emax[d * HH + hh]);
  const float ex = __expf(e[t] - m);
  e[t] = ex;
  atomicAdd(&esum[d * HH + hh], ex);
}

// ---------------------------------------------------------------------------
// Kernel 6: out[dst,h,:] += (e / (esum[dst]+eps)) * h[src,h,:]   (atomicAdd)
// ---------------------------------------------------------------------------
__global__ void gat_edge_scatter(const int* __restrict__ ei,
                                 const float* __restrict__ e,
                                 const float* __restrict__ esum,
                                 const float* __restrict__ h,
                                 float* __restrict__ out) {
  const int t = blockIdx.x * blockDim.x + threadIdx.x;
  if (t >= (EE + NN) * HH) return;
  const int i = t / HH;
  const int hh = t - i * HH;
  int s, d;
  if (i < EE) { s = ei[i]; d = ei[EE + i]; } else { s = d = i - EE; }
  const float alpha = e[t] / (esum[d * HH + hh] + 1e-16f);
  const float* __restrict__ hp = h + (size_t)s * HC + hh * CC;
  float* __restrict__ op = out + (size_t)d * HC + hh * CC;
  #pragma unroll
  for (int c = 0; c < CC; ++c) {
    atomicAdd(&op[c], alpha * hp[c]);
  }
}

// ---------------------------------------------------------------------------
extern "C" void kernel_launch(void* const* d_in, const int* in_sizes, int n_in,
                              void* d_out, int out_size, void* d_ws, size_t ws_size,
                              hipStream_t stream) {
  const float* x     = (const float*)d_in[0];
  const int*   ei    = (const int*)d_in[1];   // [2,E] row-major: src then dst
  const float* W     = (const float*)d_in[2];
  const float* a_src = (const float*)d_in[3];
  const float* a_dst = (const float*)d_in[4];
  const float* bias  = (const float*)d_in[5];
  float* out = (float*)d_out;

  // Workspace carve-out (bytes):
  //   h:    NN*HC*4          = 25,600,000
  //   as:   NN*HH*4          =  1,600,000
  //   ad:   NN*HH*4          =  1,600,000
  //   emax: NN*HH*4 (uint)   =  1,600,000
  //   esum: NN*HH*4          =  1,600,000
  //   e:    (EE+NN)*HH*4     = 20,800,000
  char* ws = (char*)d_ws;
  float*    h    = (float*)(ws);
  float*    as   = (float*)(ws + 25600000u);
  float*    ad   = (float*)(ws + 27200000u);
  unsigned* emax = (unsigned*)(ws + 28800000u);
  float*    esum = (float*)(ws + 30400000u);
  float*    e    = (float*)(ws + 32000000u);

  const int TB = 256;
  const int TE = (EE + NN) * HH;  // 5,200,000 (edge,head) work items

  gat_gemm_wmma<<<NN / 16, TB, 0, stream>>>(x, W, h);
  gat_alpha<<<(NN * HH + TB - 1) / TB, TB, 0, stream>>>(h, a_src, a_dst, as, ad);
  gat_init<<<(NN * HC + TB - 1) / TB, TB, 0, stream>>>(out, bias, emax, esum);
  gat_edge_max<<<(TE + TB - 1) / TB, TB, 0, stream>>>(ei, as, ad, e, emax);
  gat_edge_exp<<<(TE + TB - 1) / TB, TB, 0, stream>>>(ei, e, emax, esum);
  gat_edge_scatter<<<(TE + TB - 1) / TB, TB, 0, stream>>>(ei, e, esum, h, out);
}